// kMeanCluster_55714315764172
// MI455X (gfx1250) — compile-verified
//
#include <hip/hip_runtime.h>
#include <hip/hip_bf16.h>

typedef __attribute__((ext_vector_type(2))) float v2f;
typedef __attribute__((ext_vector_type(8))) float v8f;

#define DDIM   256     // feature dim (K of GEMM)
#define KCODES 512     // number of codebook entries (N of GEMM)
#define MROWS  32768   // 16 * 2048 feature rows (M of GEMM)
#define MTILE  32      // rows per block (2 A fragments per wave)
#define LDS_STRIDE 260 // 256 + 4 pad: stride%64 = 4 -> conflict-free column reads

// ---------------------------------------------------------------------------
// Kernel 1: normalize codebook rows; emit WMMA-fragment-packed cTp and csq.
// Packed layout: cTp[(k>>2)*KCODES*4 + n*4 + (k&3)] = c_norm[n][k]
//   -> a lane's B-fragment half (K = kA, kA+1 at fixed N) is 2 consecutive
//      floats => single global_load_b64 per fragment per N-tile.
// Grid: 32 blocks x 256 threads, each block handles 16 codebook rows.
// ---------------------------------------------------------------------------
__global__ __launch_bounds__(256) void normalize_codebook(
    const float* __restrict__ Ck, float* __restrict__ cTp, float* __restrict__ csq) {
  __shared__ float psum[16][17];
  __shared__ float rinv_s[16];
  const int t = threadIdx.x;
  const int rloc = t >> 4;       // local codebook row 0..15
  const int seg  = t & 15;       // 16-element segment within the row
  const int row  = blockIdx.x * 16 + rloc;   // n

  float vals[16];
  float s = 0.0f;
  const float* src = Ck + (size_t)row * DDIM + seg * 16;
#pragma unroll
  for (int j = 0; j < 16; ++j) { vals[j] = src[j]; s += vals[j] * vals[j]; }
  psum[rloc][seg] = s;
  __syncthreads();

  if (t < 16) {
    float tot = 0.0f;
#pragma unroll
    for (int j = 0; j < 16; ++j) tot += psum[t][j];
    float ri = rsqrtf(tot);
    rinv_s[t] = ri;
    csq[blockIdx.x * 16 + t] = tot * ri * ri;   // == ||c_norm||^2 (~1.0)
  }
  __syncthreads();

  const float ri = rinv_s[rloc];
#pragma unroll
  for (int j = 0; j < 16; ++j) {
    const int k = seg * 16 + j;
    cTp[(size_t)(k >> 2) * (KCODES * 4) + row * 4 + (k & 3)] = vals[j] * ri;
  }
}

// ---------------------------------------------------------------------------
// Epilogue: 32-bit C/D 16x16 layout: VGPR r, lanes 0-15: M=r, N=lane;
// lanes 16-31: M=r+8, N=lane-16.
// ---------------------------------------------------------------------------
__device__ __forceinline__ void write_tile(float* __restrict__ out,
                                           const float* __restrict__ csq,
                                           const float* fsq_s,
                                           long m0, int n, int mhalf, v8f acc) {
  const float cq = csq[n];
#pragma unroll
  for (int r = 0; r < 8; ++r) {
    const int mloc = r + mhalf;
    out[(size_t)(m0 + mloc) * KCODES + n] = fsq_s[mloc] + cq - 2.0f * acc[r];
  }
}

// ---------------------------------------------------------------------------
// Kernel 2: block = 32 feature rows x full N=512. 8 waves; wave w owns
// N in [w*64, w*64+64) with TWO A fragments (rows 0-15, 16-31) sharing each
// B fragment. Steady state / k-step: 2 ds_load_b64 + 4 global_load_b64 +
// 8 v_wmma_f32_16x16x4_f32.
// ---------------------------------------------------------------------------
__global__ __launch_bounds__(256) void dists_kernel(
    const float* __restrict__ feats, const float* __restrict__ cTp,
    const float* __restrict__ csq, float* __restrict__ out) {
  __shared__ __align__(16) float Asl[MTILE * LDS_STRIDE];
  __shared__ float psum[MTILE][9];
  __shared__ float rinv_s[MTILE];
  __shared__ float fsq_s[MTILE];

  const int t = threadIdx.x;
  const int rloc = t >> 3;          // local row 0..31 (8 threads per row)
  const int seg  = t & 7;           // 32-element segment within the row
  const long m0 = (long)blockIdx.x * MTILE;

  // --- load 32 rows (coalesced), reduce sum of squares per row ---
  float vals[32];
  float s = 0.0f;
  const float* src = feats + (size_t)(m0 + rloc) * DDIM + seg * 32;
#pragma unroll
  for (int j = 0; j < 32; ++j) { vals[j] = src[j]; s += vals[j] * vals[j]; }
  psum[rloc][seg] = s;
  __syncthreads();

  if (t < MTILE) {
    float tot = 0.0f;
#pragma unroll
    for (int j = 0; j < 8; ++j) tot += psum[t][j];
    float ri = rsqrtf(tot);
    rinv_s[t] = ri;
    fsq_s[t]  = tot * ri * ri;                  // ||f_norm||^2 (~1.0)
  }
  __syncthreads();

  {
    const float ri = rinv_s[rloc];
#pragma unroll
    for (int j = 0; j < 32; ++j)
      Asl[rloc * LDS_STRIDE + seg * 32 + j] = vals[j] * ri;
  }
  __syncthreads();

  // --- WMMA GEMM ---
  const int wave   = t >> 5;
  const int lane   = t & 31;
  const int nlane  = lane & 15;          // N (and A-row) index within fragment
  const int khalf  = (lane >> 4) * 2;    // fragment K offset: 0 or 2
  const int n_base = wave * 64;

  v8f c00 = {}, c01 = {}, c02 = {}, c03 = {};   // rows 0-15  x 4 N-tiles
  v8f c10 = {}, c11 = {}, c12 = {}, c13 = {};   // rows 16-31 x 4 N-tiles

  // B pointer in v2f units: float offset = kb*2048 + (n_base+nlane)*4 + khalf
  const v2f* __restrict__ Bv =
      (const v2f*)cTp + ((n_base + nlane) * 2 + (khalf >> 1));
  const float* Ar0 = Asl + nlane * LDS_STRIDE;           // rows 0-15
  const float* Ar1 = Ar0 + 16 * LDS_STRIDE;              // rows 16-31

#pragma unroll 2
  for (int kk = 0; kk < DDIM; kk += 4) {
    const int kA = kk + khalf;                           // even -> 8B aligned
    const v2f a0 = *(const v2f*)(Ar0 + kA);
    const v2f a1 = *(const v2f*)(Ar1 + kA);
    const v2f* B0 = Bv + (kk >> 2) * (KCODES * 2);
    const v2f b0 = B0[0];
    const v2f b1 = B0[32];
    const v2f b2 = B0[64];
    const v2f b3 = B0[96];

    c00 = __builtin_amdgcn_wmma_f32_16x16x4_f32(false, a0, false, b0, (short)0, c00, false, false);
    c10 = __builtin_amdgcn_wmma_f32_16x16x4_f32(false, a1, false, b0, (short)0, c10, false, false);
    c01 = __builtin_amdgcn_wmma_f32_16x16x4_f32(false, a0, false, b1, (short)0, c01, false, false);
    c11 = __builtin_amdgcn_wmma_f32_16x16x4_f32(false, a1, false, b1, (short)0, c11, false, false);
    c02 = __builtin_amdgcn_wmma_f32_16x16x4_f32(false, a0, false, b2, (short)0, c02, false, false);
    c12 = __builtin_amdgcn_wmma_f32_16x16x4_f32(false, a1, false, b2, (short)0, c12, false, false);
    c03 = __builtin_amdgcn_wmma_f32_16x16x4_f32(false, a0, false, b3, (short)0, c03, false, false);
    c13 = __builtin_amdgcn_wmma_f32_16x16x4_f32(false, a1, false, b3, (short)0, c13, false, false);
  }

  const int mhalf = (lane >> 4) * 8;
  write_tile(out, csq, fsq_s,      m0,      n_base + 0  + nlane, mhalf, c00);
  write_tile(out, csq, fsq_s,      m0,      n_base + 16 + nlane, mhalf, c01);
  write_tile(out, csq, fsq_s,      m0,      n_base + 32 + nlane, mhalf, c02);
  write_tile(out, csq, fsq_s,      m0,      n_base + 48 + nlane, mhalf, c03);
  write_tile(out, csq, fsq_s + 16, m0 + 16, n_base + 0  + nlane, mhalf, c10);
  write_tile(out, csq, fsq_s + 16, m0 + 16, n_base + 16 + nlane, mhalf, c11);
  write_tile(out, csq, fsq_s + 16, m0 + 16, n_base + 32 + nlane, mhalf, c12);
  write_tile(out, csq, fsq_s + 16, m0 + 16, n_base + 48 + nlane, mhalf, c13);
}

// ---------------------------------------------------------------------------
extern "C" void kernel_launch(void* const* d_in, const int* in_sizes, int n_in,
                              void* d_out, int out_size, void* d_ws, size_t ws_size,
                              hipStream_t stream) {
  const float* feats = (const float*)d_in[0];   // (16, 2048, 256) fp32
  const float* Ck    = (const float*)d_in[1];   // (1, 512, 256) fp32
  float* out = (float*)d_out;                   // (16, 2048, 512) fp32

  // workspace: cTp (256*512 floats = 512 KB, fragment-packed) | csq (512)
  float* cTp = (float*)d_ws;
  float* csq = cTp + (size_t)DDIM * KCODES;

  normalize_codebook<<<KCODES / 16, 256, 0, stream>>>(Ck, cTp, csq);
  dists_kernel<<<MROWS / MTILE, 256, 0, stream>>>(feats, cTp, csq, out);
}